// AvatarUNRES_36361193128034
// MI455X (gfx1250) — compile-verified
//
#include <hip/hip_runtime.h>
#include <math.h>

typedef float v2f __attribute__((ext_vector_type(2)));
typedef float v8f __attribute__((ext_vector_type(8)));

#define BB 2
#define NPTS 1000
#define AA 5
#define QQ 4
#define GG 800
#define CC 20                        // A*Q channels
#define KTOT 180                     // Cin(20) * taps(9)
#define GRID_ELEMS (BB*AA*QQ*GG*GG)  // 25,600,000 per branch
#define DEPTH_ELEMS (BB*AA*NPTS)     // 10,000 per branch

// output tile geometry
#define TX 16
#define TY 8
#define HW 18                        // h tile width  (TX + 2)
#define HH 10                        // h tile height (TY + 2)
#define HN (HW*HH)                   // 180 h positions
#define GW 20                        // g tile width  (TX + 4)
#define GH 12                        // g tile height (TY + 4)
#define GN (GW*GH)                   // 240 g positions

#define WS 182                       // A-matrix LDS row stride (even, !=0 mod 4)

// LDS layout in floats
#define LDS_G   0
#define LDS_H   (GN*CC)              // 4800
#define LDS_W   (LDS_H + HN*CC)      // 8400
#define LDS_TOT (LDS_W + 32*WS)      // 14224 floats = 56,896 B

// ---------------------------------------------------------------------------
// zero the per-branch count grid in workspace
__global__ void AvatarUNRES_zero(float4* __restrict__ p, int n4) {
    int i = blockIdx.x * blockDim.x + threadIdx.x;
    if (i < n4) {
        float4 z; z.x = 0.f; z.y = 0.f; z.z = 0.f; z.w = 0.f;
        p[i] = z;
    }
}

// ---------------------------------------------------------------------------
// project points: write depths, scatter counts into grid (atomic)
__global__ void AvatarUNRES_proj(const float* __restrict__ x,
                                 float* __restrict__ grid,
                                 float* __restrict__ depth_out) {
    int idx = blockIdx.x * blockDim.x + threadIdx.x;
    if (idx >= BB*AA*NPTS) return;
    int n = idx % NPTS;
    int a = (idx / NPTS) % AA;
    int b = idx / (NPTS*AA);

    const float degs[5] = {20.f, 40.f, 10.f, 15.f, 5.f};
    const float dist[5] = {20.f, 20.f, 30.f, 40.f, 80.f};
    float ang = degs[a] * 0.017453292519943295f;
    float c = cosf(ang), s = sinf(ang);

    float px = x[(b*NPTS + n)*3 + 0];
    float py = x[(b*NPTS + n)*3 + 1];
    float pz = x[(b*NPTS + n)*3 + 2];
    float rx = c*px - s*py - 0.1f;     // R(z)*p + translation
    float ry = s*px + c*py + 0.1f;
    float rz = pz + 0.1f;
    float zc = rz + dist[a];
    float u = rx / zc;                 // fx = 1
    float v = ry / zc;                 // fy = 1
    depth_out[(b*AA + a)*NPTS + n] = zc;   // fz*zc + Cz, fz=1 Cz=0

    int quad = 2*(u < 0.f) + (v < 0.f);
    int ix = (int)floorf(fabsf(u)*200.f + 1.f);   // /0.005 + Cx(=1)
    int iy = (int)floorf(fabsf(v)*200.f + 1.f);
    ix = min(max(ix, 0), GG-1);
    iy = min(max(iy, 0), GG-1);
    int cell = (((b*AA + a)*QQ + quad)*GG + iy)*GG + ix;
    atomicAdd(grid + cell, 1.0f);
}

// ---------------------------------------------------------------------------
// fused conv1 + ReLU + BN + conv2 + sigmoid, one 16x8 output tile per block.
// Implicit GEMM via V_WMMA_F32_16X16X4_F32, K permuted as k = tap*20 + cin so
// all gather offsets are compile-time constants relative to one per-lane base.
__global__ void __launch_bounds__(256)
AvatarUNRES_conv(const float* __restrict__ grid,
                 const float* __restrict__ wu, const float* __restrict__ bu,
                 const float* __restrict__ wd, const float* __restrict__ bd,
                 const float* __restrict__ bnw, const float* __restrict__ bnb,
                 float* __restrict__ out, int br) {
    __shared__ float lds[LDS_TOT];
    const int tid  = threadIdx.x;
    const int lane = tid & 31;
    const int wave = tid >> 5;
    const int b  = blockIdx.z;
    const int X0 = blockIdx.x * TX;
    const int Y0 = blockIdx.y * TY;

    // ---- phase 0: stage g tile (halo 2, zero-padded) and conv1 weights
    for (int i = tid; i < GN*CC; i += 256) {
        int cin = i / GN, rem = i % GN;
        int gy = rem / GW, gx = rem % GW;
        int iyy = Y0 - 2 + gy, ixx = X0 - 2 + gx;
        float v = 0.f;
        if (iyy >= 0 && iyy < GG && ixx >= 0 && ixx < GG)
            v = grid[((b*CC + cin)*GG + iyy)*GG + ixx];
        lds[LDS_G + cin*GN + gy*GW + gx] = v;
    }
    // A matrix [cout 0..31][k 0..179], k = tap*20 + cin (permuted im2col K)
    for (int i = tid; i < 32*KTOT; i += 256) {
        int co = i / KTOT, k = i % KTOT;
        lds[LDS_W + co*WS + k] =
            (co < CC) ? wu[(br*CC + co)*KTOT + (k % CC)*9 + (k / CC)] : 0.f;
    }
    __syncthreads();

    const float invs = 0.9999950000374997f;  // 1/sqrt(1 + 1e-5)
    const int nloc = lane & 15;
    const int kl0  = (lane < 16) ? 0 : 2;    // K split across lane halves
    const int pbA0 = LDS_W + nloc*WS + kl0;          // A base, couts 0..15
    const int pbA1 = LDS_W + (16 + nloc)*WS + kl0;   // A base, couts 16..31

    // ---- phase 1: conv1 -> h   (12 N-tiles over 180 positions, both M-tiles)
    for (int t = wave; t < 12; t += 8) {
        int ng  = t*16 + nloc;               // h position id (valid < 180)
        int ngc = (ng < HN) ? ng : (HN - 1);
        int hy = ngc / HW, hx = ngc % HW;
        int pb = LDS_G + hy*GW + hx + ((lane < 16) ? 0 : 2*GN);
        v8f acc0 = {0.f,0.f,0.f,0.f,0.f,0.f,0.f,0.f};
        v8f acc1 = {0.f,0.f,0.f,0.f,0.f,0.f,0.f,0.f};
        #pragma unroll
        for (int s = 0; s < 45; ++s) {
            const int k0  = 4*s;
            const int tap = k0 / CC, cin0 = k0 % CC;
            const int off = cin0*GN + (tap/3)*GW + (tap%3);  // compile-time
            v2f bv; bv.x = lds[pb + off]; bv.y = lds[pb + off + GN];
            v2f a0 = *(const v2f*)&lds[pbA0 + k0];
            v2f a1 = *(const v2f*)&lds[pbA1 + k0];
            acc0 = __builtin_amdgcn_wmma_f32_16x16x4_f32(
                       false, a0, false, bv, (short)0, acc0, false, false);
            acc1 = __builtin_amdgcn_wmma_f32_16x16x4_f32(
                       false, a1, false, bv, (short)0, acc1, false, false);
        }
        int imgY = Y0 - 1 + hy, imgX = X0 - 1 + hx;
        bool inb = (imgY >= 0 && imgY < GG && imgX >= 0 && imgX < GG);
        #pragma unroll
        for (int r = 0; r < 8; ++r) {
            int c0 = r + ((lane < 16) ? 0 : 8);      // acc0 -> couts 0..15
            if (ng < HN) {
                float z = acc0[r] + bu[br*CC + c0];
                lds[LDS_H + c0*HN + ng] =
                    inb ? (fmaxf(z, 0.f)*invs*bnw[c0] + bnb[c0]) : 0.f;
                int c1 = 16 + c0;                     // acc1 -> couts 16..19
                if (c1 < CC) {
                    float z1 = acc1[r] + bu[br*CC + c1];
                    lds[LDS_H + c1*HN + ng] =
                        inb ? (fmaxf(z1, 0.f)*invs*bnw[c1] + bnb[c1]) : 0.f;
                }
            }
        }
    }
    __syncthreads();

    // ---- swap in conv2 weights (reuse A buffer, same K permutation)
    for (int i = tid; i < 32*KTOT; i += 256) {
        int co = i / KTOT, k = i % KTOT;
        lds[LDS_W + co*WS + k] =
            (co < CC) ? wd[(br*CC + co)*KTOT + (k % CC)*9 + (k / CC)] : 0.f;
    }
    __syncthreads();

    // ---- phase 2: conv2 -> sigmoid -> out (8 N-tiles == 8 waves)
    {
        int oy = wave, ox = nloc;            // 8 rows x 16 cols
        int pb = LDS_H + oy*HW + ox + ((lane < 16) ? 0 : 2*HN);
        v8f acc0 = {0.f,0.f,0.f,0.f,0.f,0.f,0.f,0.f};
        v8f acc1 = {0.f,0.f,0.f,0.f,0.f,0.f,0.f,0.f};
        #pragma unroll
        for (int s = 0; s < 45; ++s) {
            const int k0  = 4*s;
            const int tap = k0 / CC, cin0 = k0 % CC;
            const int off = cin0*HN + (tap/3)*HW + (tap%3);  // compile-time
            v2f bv; bv.x = lds[pb + off]; bv.y = lds[pb + off + HN];
            v2f a0 = *(const v2f*)&lds[pbA0 + k0];
            v2f a1 = *(const v2f*)&lds[pbA1 + k0];
            acc0 = __builtin_amdgcn_wmma_f32_16x16x4_f32(
                       false, a0, false, bv, (short)0, acc0, false, false);
            acc1 = __builtin_amdgcn_wmma_f32_16x16x4_f32(
                       false, a1, false, bv, (short)0, acc1, false, false);
        }
        #pragma unroll
        for (int r = 0; r < 8; ++r) {
            int c0 = r + ((lane < 16) ? 0 : 8);
            float z0 = acc0[r] + bd[br*CC + c0];
            out[((b*CC + c0)*GG + (Y0 + oy))*GG + (X0 + ox)] =
                1.f / (1.f + expf(-z0));
            int c1 = 16 + c0;
            if (c1 < CC) {
                float z1 = acc1[r] + bd[br*CC + c1];
                out[((b*CC + c1)*GG + (Y0 + oy))*GG + (X0 + ox)] =
                    1.f / (1.f + expf(-z1));
            }
        }
    }
}

// ---------------------------------------------------------------------------
extern "C" void kernel_launch(void* const* d_in, const int* in_sizes, int n_in,
                              void* d_out, int out_size, void* d_ws, size_t ws_size,
                              hipStream_t stream) {
    (void)in_sizes; (void)n_in; (void)out_size; (void)ws_size;
    const float* xs[4] = { (const float*)d_in[1], (const float*)d_in[2],
                           (const float*)d_in[3], (const float*)d_in[4] };
    const float* wu  = (const float*)d_in[5];
    const float* bu  = (const float*)d_in[6];
    const float* wd  = (const float*)d_in[7];
    const float* bd  = (const float*)d_in[8];
    const float* bnw = (const float*)d_in[9];
    const float* bnb = (const float*)d_in[10];
    float* out  = (float*)d_out;
    float* grid = (float*)d_ws;              // one branch's count grid (102.4 MB)

    const int n4 = GRID_ELEMS / 4;
    for (int br = 0; br < 4; ++br) {
        AvatarUNRES_zero<<<(n4 + 255)/256, 256, 0, stream>>>((float4*)d_ws, n4);
        AvatarUNRES_proj<<<(BB*AA*NPTS + 255)/256, 256, 0, stream>>>(
            xs[br], grid,
            out + (size_t)4*GRID_ELEMS + (size_t)br*DEPTH_ELEMS);
        dim3 g(GG/TX, GG/TY, BB);
        AvatarUNRES_conv<<<g, 256, 0, stream>>>(
            grid, wu, bu, wd, bd, bnw, bnb,
            out + (size_t)br*GRID_ELEMS, br);
    }
}